// PointsetGrouper_formal_78271484002329
// MI455X (gfx1250) — compile-verified
//
#include <hip/hip_runtime.h>
#include <stdint.h>

// Problem constants (fixed by the reference harness)
#define B_  4
#define N_  8192
#define C_  256
#define S_  2048      // N / REDUCE
#define NS_ 32        // NSAMPLE
#define R2_ 0.04f     // RADIUS^2

typedef __attribute__((ext_vector_type(2))) float v2f;
typedef __attribute__((ext_vector_type(8))) float v8f;

// ---------------- wave32 xor-shuffle via ds_swizzle (group-of-32, and=0x1f) ---
template<int XOR>
__device__ __forceinline__ float swzf(float v) {
  return __int_as_float(__builtin_amdgcn_ds_swizzle(__float_as_int(v), (XOR << 10) | 0x1f));
}
template<int XOR>
__device__ __forceinline__ int swzi(int v) {
  return __builtin_amdgcn_ds_swizzle(v, (XOR << 10) | 0x1f);
}

template<int XOR>
__device__ __forceinline__ void argmax_step(float& d, int& i) {
  float od = swzf<XOR>(d);
  int   oi = swzi<XOR>(i);
  // larger dist wins; ties -> smaller index (matches jnp.argmax first-occurrence)
  if (od > d || (od == d && oi < i)) { d = od; i = oi; }
}

__device__ __forceinline__ void wave_argmax(float& d, int& i) {
  argmax_step<16>(d, i);
  argmax_step<8>(d, i);
  argmax_step<4>(d, i);
  argmax_step<2>(d, i);
  argmax_step<1>(d, i);
}

// ---------------- Kernel 1: points (B,C,N) -> ptsT (B,N,C) ------------------
__global__ __launch_bounds__(256)
void transpose_kernel(const float* __restrict__ pts, float* __restrict__ ptsT) {
  __shared__ float tile[32][33];
  const int b  = blockIdx.z;
  const int n0 = blockIdx.x * 32;
  const int c0 = blockIdx.y * 32;
  const int tx = threadIdx.x;   // 0..31 (along N for read)
  const int ty = threadIdx.y;   // 0..7
  const float* src = pts + ((size_t)b * C_ + c0) * N_ + n0;
  #pragma unroll
  for (int i = 0; i < 4; ++i)
    tile[ty + 8 * i][tx] = src[(size_t)(ty + 8 * i) * N_ + tx];
  __syncthreads();
  float* dst = ptsT + ((size_t)b * N_ + n0) * C_ + c0;
  #pragma unroll
  for (int i = 0; i < 4; ++i)
    dst[(size_t)(ty + 8 * i) * C_ + tx] = tile[tx][ty + 8 * i];
}

// ---------------- Kernel 2: furthest point sampling -------------------------
// One block per batch; coords + running min-dists live in registers
// (8 points / thread). 2047 sequential argmax steps (inherent to FPS).
__global__ __launch_bounds__(1024)
void fps_kernel(const float* __restrict__ xyz, int* __restrict__ fps_idx,
                float* __restrict__ new_xyz) {
  const int b   = blockIdx.x;
  const int tid = threadIdx.x;
  const float* xb = xyz + (size_t)b * N_ * 3;

  __shared__ float sd[32];
  __shared__ int   si[32];
  __shared__ int   sbest;

  float dists[8], px[8], py[8], pz[8];
  #pragma unroll
  for (int j = 0; j < 8; ++j) {
    const int n = tid + j * 1024;
    dists[j] = 1e10f;
    px[j] = xb[n * 3 + 0];
    py[j] = xb[n * 3 + 1];
    pz[j] = xb[n * 3 + 2];
  }

  if (tid == 0) {
    fps_idx[b * S_] = 0;
    new_xyz[((size_t)b * S_) * 3 + 0] = xb[0];
    new_xyz[((size_t)b * S_) * 3 + 1] = xb[1];
    new_xyz[((size_t)b * S_) * 3 + 2] = xb[2];
  }

  float lx = xb[0], ly = xb[1], lz = xb[2];   // broadcast loads (uniform addr)

  for (int s = 1; s < S_; ++s) {
    float bd = -1.0f; int bi = 0x7fffffff;
    #pragma unroll
    for (int j = 0; j < 8; ++j) {
      const float dx = px[j] - lx, dy = py[j] - ly, dz = pz[j] - lz;
      const float d  = dx * dx + dy * dy + dz * dz;
      const float m  = dists[j] < d ? dists[j] : d;
      dists[j] = m;
      const int n = tid + j * 1024;
      if (m > bd || (m == bd && n < bi)) { bd = m; bi = n; }
    }
    wave_argmax(bd, bi);
    if ((tid & 31) == 0) { sd[tid >> 5] = bd; si[tid >> 5] = bi; }
    __syncthreads();
    if (tid < 32) {
      float bd2 = sd[tid]; int bi2 = si[tid];
      wave_argmax(bd2, bi2);
      if (tid == 0) {
        sbest = bi2;
        fps_idx[b * S_ + s] = bi2;
        new_xyz[((size_t)b * S_ + s) * 3 + 0] = xb[bi2 * 3 + 0];
        new_xyz[((size_t)b * S_ + s) * 3 + 1] = xb[bi2 * 3 + 1];
        new_xyz[((size_t)b * S_ + s) * 3 + 2] = xb[bi2 * 3 + 2];
      }
    }
    __syncthreads();
    const int nl = sbest;
    lx = xb[nl * 3 + 0]; ly = xb[nl * 3 + 1]; lz = xb[nl * 3 + 2];
  }
}

// ---------------- Kernel 3: ball query via V_WMMA_F32_16X16X4_F32 -----------
// One wave handles a tile of 16 query points. Per 16-support tile:
//   D[m][n] = |s_n|^2 - 2 q_m . s_n   (one f32 WMMA, K=4 trick)
//   valid   = D[m][n] < r^2 - |q_m|^2
// D VGPR k holds rows {k, k+8} across the two lane halves, so one ballot per
// VGPR produces two 16-bit per-row masks; per-lane ctz loop appends indices
// in ascending order (matches pointnet2 ball_query semantics).
__global__ __launch_bounds__(32)
void ballq_kernel(const float* __restrict__ xyz, const float* __restrict__ new_xyz,
                  int* __restrict__ idx_out) {
  const int tilesPerB = S_ / 16;
  const int b    = blockIdx.x / tilesPerB;
  const int tile = blockIdx.x % tilesPerB;
  const int l    = threadIdx.x;
  const int m    = l & 15;
  const int qs   = tile * 16 + m;

  const float* xb = xyz + (size_t)b * N_ * 3;
  const float* qp = new_xyz + ((size_t)b * S_ + qs) * 3;
  const float qx = qp[0], qy = qp[1], qz = qp[2];

  __shared__ float tsh[16];
  if (l < 16) tsh[m] = R2_ - (qx * qx + qy * qy + qz * qz);
  __syncthreads();

  // A (16x4 f32): lane m holds K0,K1; lane m+16 holds K2,K3
  v2f a;
  if (l < 16) { a.x = -2.0f * qx; a.y = -2.0f * qy; }
  else        { a.x = -2.0f * qz; a.y = 1.0f; }

  float th[8];
  #pragma unroll
  for (int k = 0; k < 8; ++k) th[k] = tsh[k + (l >= 16 ? 8 : 0)];

  int cnt = 0, first = 0;
  const size_t qbase = ((size_t)b * S_ + qs) * NS_;

  for (int n0 = 0; n0 < N_; n0 += 16) {
    const int n = n0 + m;
    const float* sp = xb + (size_t)n * 3;
    const float sx = sp[0], sy = sp[1], sz = sp[2];
    // B (4x16 f32): lanes 0-15 hold rows K0,K1; lanes 16-31 hold rows K2,K3
    v2f bb;
    if (l < 16) { bb.x = sx; bb.y = sy; }
    else        { bb.x = sz; bb.y = sx * sx + sy * sy + sz * sz; }

    v8f acc = {};
    acc = __builtin_amdgcn_wmma_f32_16x16x4_f32(
        /*neg_a=*/false, a, /*neg_b=*/false, bb,
        /*c_mod=*/(short)0, acc, /*reuse_a=*/false, /*reuse_b=*/false);

    unsigned bal[8];
    #pragma unroll
    for (int k = 0; k < 8; ++k)
      bal[k] = __builtin_amdgcn_ballot_w32(acc[k] < th[k]);

    if (l < 16) {
      // row r mask: r<8 -> bal[r] low16 ; r>=8 -> bal[r-8] high16
      unsigned sel = bal[0];
      const unsigned rr = (unsigned)(l & 7);
      #pragma unroll
      for (int k = 1; k < 8; ++k) sel = (rr == (unsigned)k) ? bal[k] : sel;
      unsigned mask = (l < 8) ? (sel & 0xffffu) : (sel >> 16);
      while (mask && cnt < NS_) {
        const int nn = n0 + __builtin_ctz(mask);
        if (cnt == 0) first = nn;
        idx_out[qbase + cnt] = nn;
        ++cnt;
        mask &= mask - 1;
      }
    }
    const unsigned done = __builtin_amdgcn_ballot_w32(l >= 16 || cnt >= NS_);
    if (done == 0xffffffffu) break;   // uniform early exit (EXEC stays full)
  }

  if (l < 16) {
    const int fill = (cnt > 0) ? first : 0;   // argmax(all-false)==0 semantics
    for (int k = cnt; k < NS_; ++k) idx_out[qbase + k] = fill;
  }
}

// ---------------- Kernel 4: gather + mean + affine ---------------------------
// One block per (b,s); thread = channel. Reads contiguous 1 KB rows of ptsT
// (L2-resident, coalesced), keeps the 32 gathered values in registers, writes
// the 268 MB output fully coalesced.
__global__ __launch_bounds__(256)
void group_kernel(const float* __restrict__ ptsT, const int* __restrict__ idx,
                  const float* __restrict__ alpha, const float* __restrict__ beta,
                  float* __restrict__ outg) {
  const int bs = blockIdx.x;          // b*S + s
  const int b  = bs / S_;
  const int c  = threadIdx.x;         // 0..C-1
  __shared__ int gidx[NS_];
  if (c < NS_) gidx[c] = idx[(size_t)bs * NS_ + c];
  __syncthreads();
  const float* base = ptsT + (size_t)b * N_ * C_;
  float v[NS_];
  float sum = 0.0f;
  #pragma unroll
  for (int k = 0; k < NS_; ++k) {
    v[k] = base[(size_t)gidx[k] * C_ + c];
    sum += v[k];
  }
  const float mean = sum * (1.0f / NS_);
  const float al = alpha[c], be = beta[c];
  float* o = outg + (size_t)bs * NS_ * C_ + c;
  #pragma unroll
  for (int k = 0; k < NS_; ++k)
    o[(size_t)k * C_] = al * (v[k] - mean) + be;
}

// ---------------- launch -----------------------------------------------------
extern "C" void kernel_launch(void* const* d_in, const int* in_sizes, int n_in,
                              void* d_out, int out_size, void* d_ws, size_t ws_size,
                              hipStream_t stream) {
  const float* xyz    = (const float*)d_in[0];   // (B,N,3)
  const float* points = (const float*)d_in[1];   // (B,C,N)
  const float* alpha  = (const float*)d_in[2];   // (C)
  const float* beta   = (const float*)d_in[3];   // (C)

  float* out      = (float*)d_out;
  float* new_xyz  = out;                          // (B,S,3)
  float* grouped  = out + (size_t)B_ * S_ * 3;    // (B,S,NS,C)

  char* ws = (char*)d_ws;
  int* fps_idx  = (int*)ws;  ws += sizeof(int) * (size_t)B_ * S_;        // 32 KB
  int* ball_idx = (int*)ws;  ws += sizeof(int) * (size_t)B_ * S_ * NS_;  // 1 MB
  float* ptsT   = (float*)ws;                                            // 32 MB

  transpose_kernel<<<dim3(N_ / 32, C_ / 32, B_), dim3(32, 8), 0, stream>>>(points, ptsT);
  fps_kernel<<<B_, 1024, 0, stream>>>(xyz, fps_idx, new_xyz);
  ballq_kernel<<<B_ * (S_ / 16), 32, 0, stream>>>(xyz, new_xyz, ball_idx);
  group_kernel<<<B_ * S_, C_, 0, stream>>>(ptsT, ball_idx, alpha, beta, grouped);
}